// GATv2_23278722744604
// MI455X (gfx1250) — compile-verified
//
#include <hip/hip_runtime.h>

// ---------------- problem constants (match reference) ----------------
#define N_NODES 50000
#define N_EDGES 800000
#define FEATS   128          // = HEADS * DOUT
#define HEADS   4
#define DOUT    32
#define SLOPE   0.2f
#define TS      130          // padded LDS row stride (dwords): avoids 16-way bank conflicts, keeps 8B align

typedef __attribute__((ext_vector_type(2))) float v2f;
typedef __attribute__((ext_vector_type(8))) float v8f;

// Monotone float <-> uint encoding so segment-max can use integer atomicMax.
__device__ __forceinline__ unsigned fenc(float f) {
    unsigned u = __float_as_uint(f);
    return (u & 0x80000000u) ? ~u : (u | 0x80000000u);
}
__device__ __forceinline__ float fdec(unsigned u) {
    return __uint_as_float((u & 0x80000000u) ? (u & 0x7fffffffu) : ~u);
}
__device__ __forceinline__ float elu1(float x) {
    return x > 0.f ? x : (__expf(x) - 1.f);
}

// ---------------------------------------------------------------------
// Projection: fsrc = h @ W_src + b_src ; fdst = h @ W_dst + b_dst
// grid = (N/16, 4); block = 128 (4 waves).
// blockIdx.y: bit1 = matrix (0=src,1=dst), bit0 = which half of 8 col-tiles.
// Each wave computes one 16x16 output tile with 32 chained
// v_wmma_f32_16x16x4_f32 ops, operands streamed from LDS.
// ---------------------------------------------------------------------
__global__ __launch_bounds__(128) void proj_kernel(
    const float* __restrict__ h,
    const float* __restrict__ Wsrc, const float* __restrict__ bsrc,
    const float* __restrict__ Wdst, const float* __restrict__ bdst,
    float* __restrict__ fsrc, float* __restrict__ fdst)
{
    __shared__ float ldsW[4 * 16 * TS];   // 4 col-tiles, transposed: [tile][n (16)][k (128, pad->130)]
    __shared__ float ldsA[16 * TS];       // 16 rows of A: [r][k]

    const int tid  = threadIdx.x;
    const int row0 = blockIdx.x * 16;
    const int mat  = blockIdx.y >> 1;         // 0 = src, 1 = dst
    const int chal = blockIdx.y & 1;          // column half (tiles 0..3 or 4..7)
    const float* __restrict__ W    = mat ? Wdst : Wsrc;
    const float* __restrict__ bias = mat ? bdst : bsrc;
    float* __restrict__ out        = mat ? fdst : fsrc;

    // Stage W transposed: 4 tiles x 16 cols x 128 k.  i enumerates (k, t, nl)
    // so the global read of W[k*128 + n] is coalesced across lanes.
    for (int i = tid; i < 4 * 16 * 128; i += 128) {
        int k  = i >> 6;
        int t  = (i >> 4) & 3;
        int nl = i & 15;
        int n  = chal * 64 + t * 16 + nl;
        ldsW[t * (16 * TS) + nl * TS + k] = W[k * 128 + n];
    }
    // Stage 16 rows of h (A operand), coalesced.
    for (int i = tid; i < 16 * 128; i += 128) {
        int r = i >> 7;
        int k = i & 127;
        ldsA[r * TS + k] = h[(row0 + r) * 128 + k];
    }
    __syncthreads();

    const int wave = tid >> 5;            // 0..3 -> local col tile
    const int lane = tid & 31;
    const int nr   = lane & 15;           // A: M row / B: N col held by this lane
    const int koff = (lane >> 4) << 1;    // lanes 0-15 hold K{0,1}, lanes 16-31 hold K{2,3}

    const float* wt = &ldsW[wave * (16 * TS) + nr * TS];
    const float* at = &ldsA[nr * TS];

    v8f acc = {};
#pragma unroll
    for (int k0 = 0; k0 < 128; k0 += 4) {
        v2f a = *(const v2f*)(at + k0 + koff);   // A 16x4 f32 (ISA layout)
        v2f b = *(const v2f*)(wt + k0 + koff);   // B 4x16 f32 (mirrored layout)
        acc = __builtin_amdgcn_wmma_f32_16x16x4_f32(
            /*neg_a=*/false, a, /*neg_b=*/false, b,
            /*c_mod=*/(short)0, acc, /*reuse_a=*/false, /*reuse_b=*/false);
    }

    const int col   = chal * 64 + wave * 16 + nr;
    const float bv  = bias[col];
    const int rbase = row0 + ((lane >> 4) << 3);   // lanes 16-31 cover M=8..15
#pragma unroll
    for (int g = 0; g < 8; ++g)
        out[(rbase + g) * 128 + col] = acc[g] + bv;
}

// ---------------- segment-state init: m_enc = 0 (< enc(-inf)), den = 0 --------
__global__ void init_md_kernel(unsigned* __restrict__ m_enc, float* __restrict__ den)
{
    int i = blockIdx.x * blockDim.x + threadIdx.x;
    if (i < N_NODES * HEADS) { m_enc[i] = 0u; den[i] = 0.f; }
}

// ---------------- pass A: per-(edge,head) GATv2 logit + segment max ---------
__global__ void edge_logits_kernel(
    const float* __restrict__ fsrc, const float* __restrict__ fdst,
    const int* __restrict__ src, const int* __restrict__ dst,
    const float* __restrict__ attn,
    float* __restrict__ logits, unsigned* __restrict__ m_enc)
{
    int tid = blockIdx.x * blockDim.x + threadIdx.x;
    if (tid >= N_EDGES * HEADS) return;
    int e  = tid >> 2;
    int hd = tid & 3;
    const float* ps = fsrc + src[e] * 128 + hd * 32;
    const float* pd = fdst + dst[e] * 128 + hd * 32;
    const float* pa = attn + hd * 32;
    float s = 0.f;
#pragma unroll
    for (int d = 0; d < 32; d += 4) {
        float4 a = *(const float4*)(ps + d);
        float4 b = *(const float4*)(pd + d);
        float4 w = *(const float4*)(pa + d);
        float x;
        x = a.x + b.x; s += w.x * (x > 0.f ? x : SLOPE * x);
        x = a.y + b.y; s += w.y * (x > 0.f ? x : SLOPE * x);
        x = a.z + b.z; s += w.z * (x > 0.f ? x : SLOPE * x);
        x = a.w + b.w; s += w.w * (x > 0.f ? x : SLOPE * x);
    }
    logits[tid] = s;
    atomicMax(&m_enc[dst[e] * HEADS + hd], fenc(s));
}

// ---------------- pass B: ex = exp(logit - m[dst]); den[dst] += ex ----------
__global__ void edge_exp_kernel(
    const int* __restrict__ dst,
    float* __restrict__ logits, const unsigned* __restrict__ m_enc,
    float* __restrict__ den)
{
    int tid = blockIdx.x * blockDim.x + threadIdx.x;
    if (tid >= N_EDGES * HEADS) return;
    int e  = tid >> 2;
    int hd = tid & 3;
    int dn = dst[e] * HEADS + hd;
    float ex = __expf(logits[tid] - fdec(m_enc[dn]));
    logits[tid] = ex;
    atomicAdd(&den[dn], ex);
}

// ---------------- out init: residual (identity) + bias ----------------------
__global__ void init_out_kernel(const float* __restrict__ hin,
                                const float* __restrict__ bias,
                                float* __restrict__ out)
{
    int i = blockIdx.x * blockDim.x + threadIdx.x;
    if (i < N_NODES * 128) out[i] = hin[i] + bias[i & 127];
}

// ---------------- pass C: out[dst] += (ex/den) * fsrc[src] ------------------
__global__ void edge_scatter_kernel(
    const float* __restrict__ fsrc,
    const int* __restrict__ src, const int* __restrict__ dst,
    const float* __restrict__ ex, const float* __restrict__ den,
    float* __restrict__ out)
{
    int tid = blockIdx.x * blockDim.x + threadIdx.x;
    if (tid >= N_EDGES * HEADS) return;
    int e  = tid >> 2;
    int hd = tid & 3;
    int dn = dst[e];
    float dv = den[dn * HEADS + hd];
    float w  = ex[tid] / (dv > 0.f ? dv : 1.f);
    const float* ps = fsrc + src[e] * 128 + hd * 32;
    float* po       = out + dn * 128 + hd * 32;
#pragma unroll
    for (int d = 0; d < 32; ++d)
        atomicAdd(po + d, w * ps[d]);
}

// ---------------- layer-1 activation: elu applied twice ---------------------
__global__ void elu2_kernel(float* __restrict__ x)
{
    int i = blockIdx.x * blockDim.x + threadIdx.x;
    if (i < N_NODES * 128) x[i] = elu1(elu1(x[i]));
}

// ---------------- final: mean over heads -> [N, 32] -------------------------
__global__ void mean_kernel(const float* __restrict__ h, float* __restrict__ out)
{
    int i = blockIdx.x * blockDim.x + threadIdx.x;
    if (i >= N_NODES * DOUT) return;
    int n = i >> 5, d = i & 31;
    const float* p = h + n * 128 + d;
    out[i] = 0.25f * (p[0] + p[32] + p[64] + p[96]);
}

// ---------------------------------------------------------------------------
extern "C" void kernel_launch(void* const* d_in, const int* in_sizes, int n_in,
                              void* d_out, int out_size, void* d_ws, size_t ws_size,
                              hipStream_t stream)
{
    const float* feats = (const float*)d_in[0];
    const int*   src   = (const int*)d_in[1];
    const int*   dst   = (const int*)d_in[2];
    const float* Wsrc0 = (const float*)d_in[3];
    const float* bsrc0 = (const float*)d_in[4];
    const float* Wdst0 = (const float*)d_in[5];
    const float* bdst0 = (const float*)d_in[6];
    const float* attn0 = (const float*)d_in[7];
    const float* bias0 = (const float*)d_in[8];
    const float* Wsrc1 = (const float*)d_in[9];
    const float* bsrc1 = (const float*)d_in[10];
    const float* Wdst1 = (const float*)d_in[11];
    const float* Wdst1b= (const float*)d_in[12];
    const float* attn1 = (const float*)d_in[13];
    const float* bias1 = (const float*)d_in[14];
    const float* bdst1 = Wdst1b;
    float* outp = (float*)d_out;

    // workspace layout (floats)
    float*    ws    = (float*)d_ws;
    float*    fsrc  = ws;                         // N*128
    float*    fdst  = ws + 6400000;               // N*128 (reused as layer-2 out)
    float*    bufC  = ws + 12800000;              // N*128 (layer-1 out / layer-2 in)
    float*    ex    = ws + 19200000;              // E*H
    unsigned* m_enc = (unsigned*)(ws + 22400000); // N*H
    float*    den   = ws + 22600000;              // N*H

    const dim3 projGrid(N_NODES / 16, 4);
    const int  EH_BLK = (N_EDGES * HEADS + 255) / 256;
    const int  NF_BLK = (N_NODES * 128 + 255) / 256;
    const int  NH_BLK = (N_NODES * HEADS + 255) / 256;

    // ---------------- Layer 1 ----------------
    init_md_kernel<<<NH_BLK, 256, 0, stream>>>(m_enc, den);
    proj_kernel<<<projGrid, 128, 0, stream>>>(feats, Wsrc0, bsrc0, Wdst0, bdst0, fsrc, fdst);
    edge_logits_kernel<<<EH_BLK, 256, 0, stream>>>(fsrc, fdst, src, dst, attn0, ex, m_enc);
    edge_exp_kernel<<<EH_BLK, 256, 0, stream>>>(dst, ex, m_enc, den);
    init_out_kernel<<<NF_BLK, 256, 0, stream>>>(feats, bias0, bufC);
    edge_scatter_kernel<<<EH_BLK, 256, 0, stream>>>(fsrc, src, dst, ex, den, bufC);
    elu2_kernel<<<NF_BLK, 256, 0, stream>>>(bufC);

    // ---------------- Layer 2 ----------------
    init_md_kernel<<<NH_BLK, 256, 0, stream>>>(m_enc, den);
    proj_kernel<<<projGrid, 128, 0, stream>>>(bufC, Wsrc1, bsrc1, Wdst1, bdst1, fsrc, fdst);
    edge_logits_kernel<<<EH_BLK, 256, 0, stream>>>(fsrc, fdst, src, dst, attn1, ex, m_enc);
    edge_exp_kernel<<<EH_BLK, 256, 0, stream>>>(dst, ex, m_enc, den);
    init_out_kernel<<<NF_BLK, 256, 0, stream>>>(bufC, bias1, fdst);   // fdst free after pass A
    edge_scatter_kernel<<<EH_BLK, 256, 0, stream>>>(fsrc, src, dst, ex, den, fdst);
    mean_kernel<<<(N_NODES * DOUT + 255) / 256, 256, 0, stream>>>(fdst, outp);
}